// AssimilationLoss_31018253812261
// MI455X (gfx1250) — compile-verified
//
#include <hip/hip_runtime.h>
#include <hip/hip_bf16.h>

#ifndef __has_builtin
#define __has_builtin(x) 0
#endif

#define HAS_ASYNC_LDS  __has_builtin(__builtin_amdgcn_global_load_async_to_lds_b128)
#define HAS_ASYNC_WAIT __has_builtin(__builtin_amdgcn_s_wait_asynccnt)
#define HAS_WMMA_F32X4 __has_builtin(__builtin_amdgcn_wmma_f32_16x16x4_f32)

typedef __attribute__((ext_vector_type(2))) float v2f;
typedef __attribute__((ext_vector_type(8))) float v8f;
typedef __attribute__((ext_vector_type(4))) int   v4i;

static constexpr int   kB                = 8;
static constexpr int   kN                = 8192;
static constexpr int   kD                = 512;
static constexpr int   kBlocksPerBatch   = 64;
static constexpr int   kRowsPerBlock     = kN / kBlocksPerBatch;   // 128
static constexpr int   kWaves            = 8;
static constexpr int   kRowsPerWave      = kRowsPerBlock / kWaves; // 16
static constexpr float kEps              = 1e-8f;

// ---- async-counter waits (CDNA5: split counters, loads complete in order) ----
__device__ __forceinline__ void wait_async_le4() {
#if HAS_ASYNC_WAIT
  __builtin_amdgcn_s_wait_asynccnt(4);
#elif defined(__gfx1250__)
  asm volatile("s_wait_asynccnt 4" ::: "memory");
#endif
  asm volatile("" ::: "memory");  // compiler ordering fence for LDS reads
}
__device__ __forceinline__ void wait_async_le0() {
#if HAS_ASYNC_WAIT
  __builtin_amdgcn_s_wait_asynccnt(0);
#elif defined(__gfx1250__)
  asm volatile("s_wait_asynccnt 0" ::: "memory");
#endif
  asm volatile("" ::: "memory");
}

// =====================================================================
// Kernel 1: stream x once. One wave per row: sum-of-squares via wave32
// shfl butterfly, scale row by 1/max(||row||, eps), accumulate into
// per-lane registers. Deterministic fixed-order cross-wave combine in
// LDS, partial s-vectors to workspace.  Rows are double-buffered through
// LDS via GLOBAL_LOAD_ASYNC_TO_LDS_B128 when available.
// =====================================================================
__global__ __launch_bounds__(256) void
assim_partials(const float* __restrict__ x, float* __restrict__ partial)
{
  const int lane  = threadIdx.x & 31;
  const int wave  = threadIdx.x >> 5;
  const int blk   = blockIdx.x;   // 0..63
  const int batch = blockIdx.y;   // 0..7

  const size_t rowBase =
      (size_t)batch * kN + (size_t)blk * kRowsPerBlock + (size_t)wave * kRowsPerWave;
  // lane owns elements d = c*128 + lane*4 + j  (c=0..3, j=0..3) — coalesced float4s
  const float* src = x + rowBase * kD + lane * 4;

  __shared__ float stage[kWaves][2][kD];   // 32 KB async double-buffer
  __shared__ float waveAcc[kWaves][kD];    // 16 KB combine buffer

  float4 acc[4];
#pragma unroll
  for (int c = 0; c < 4; ++c) acc[c] = make_float4(0.f, 0.f, 0.f, 0.f);

#if HAS_ASYNC_LDS
  float* myStage = &stage[wave][0][0];
  auto issueRow = [&](int i, int buf) {
    const float* g = src + (size_t)i * kD;
    float*       l = myStage + buf * kD + lane * 4;
#pragma unroll
    for (int c = 0; c < 4; ++c) {
      __builtin_amdgcn_global_load_async_to_lds_b128(
          (__attribute__((address_space(1))) v4i*)(g + c * 128),
          (__attribute__((address_space(3))) v4i*)(l + c * 128),
          0, 0);
    }
  };
  issueRow(0, 0);
#endif

  for (int i = 0; i < kRowsPerWave; ++i) {
    float4 v[4];
#if HAS_ASYNC_LDS
    if (i + 1 < kRowsPerWave) { issueRow(i + 1, (i + 1) & 1); wait_async_le4(); }
    else                      { wait_async_le0(); }
    const float* row = myStage + (i & 1) * kD + lane * 4;
#pragma unroll
    for (int c = 0; c < 4; ++c) v[c] = *(const float4*)(row + c * 128);
#else
    const float* row = src + (size_t)i * kD;
#pragma unroll
    for (int c = 0; c < 4; ++c) v[c] = *(const float4*)(row + c * 128);
#endif

    // row sum of squares (16 per-lane + 5-step wave32 butterfly)
    float ss = 0.f;
#pragma unroll
    for (int c = 0; c < 4; ++c)
      ss += v[c].x * v[c].x + v[c].y * v[c].y + v[c].z * v[c].z + v[c].w * v[c].w;
#pragma unroll
    for (int off = 16; off > 0; off >>= 1) ss += __shfl_xor(ss, off, 32);

    const float inv = 1.0f / fmaxf(sqrtf(ss), kEps);
#pragma unroll
    for (int c = 0; c < 4; ++c) {
      acc[c].x += v[c].x * inv; acc[c].y += v[c].y * inv;
      acc[c].z += v[c].z * inv; acc[c].w += v[c].w * inv;
    }
  }

  // deterministic fixed-order combine of the 8 wave accumulators
#pragma unroll
  for (int c = 0; c < 4; ++c)
    *(float4*)&waveAcc[wave][c * 128 + lane * 4] = acc[c];
  __syncthreads();

  float* dst = partial + ((size_t)batch * kBlocksPerBatch + blk) * kD;
  for (int d = threadIdx.x; d < kD; d += 256) {
    float s = 0.f;
#pragma unroll
    for (int w = 0; w < kWaves; ++w) s += waveAcc[w][d];
    dst[d] = s;
  }
}

// =====================================================================
// Kernel 2: reduce the 64 partials per batch into S[16][512] (rows 8..15
// zero), then Gram diagonal via v_wmma_f32_16x16x4_f32 (A == B == S
// fragment, since Gram = S*S^T), scale by 1/(N*N*B), write scalar.
// =====================================================================
__global__ __launch_bounds__(256) void
assim_finalize(const float* __restrict__ partial, float* __restrict__ out)
{
  __shared__ float S[16][kD];   // 32 KB, batches padded to 16 for WMMA
  __shared__ float diag[16];

  for (int idx = threadIdx.x; idx < 16 * kD; idx += 256) {
    const int b = idx >> 9;        // / 512
    const int d = idx & (kD - 1);
    float s = 0.f;
    if (b < kB) {
      const float* p = partial + (size_t)b * kBlocksPerBatch * kD + d;
      for (int k = 0; k < kBlocksPerBatch; ++k) s += p[(size_t)k * kD];
    }
    S[b][d] = s;
  }
  __syncthreads();

  const int lane = threadIdx.x & 31;
  const int wave = threadIdx.x >> 5;

  if (wave == 0) {  // uniform branch: EXEC all-ones within the wave for WMMA
#if HAS_WMMA_F32X4
    v8f c = {0.f, 0.f, 0.f, 0.f, 0.f, 0.f, 0.f, 0.f};
    // A 16x4 f32 layout: lanes 0-15 = rows M, K0/K1 in v0/v1; lanes 16-31 = K2/K3.
    // B 4x16 layout holds B[k][n] = S[n][k] at the same (lane, reg) slots -> reuse A.
    const int m    = lane & 15;
    const int koff = (lane >> 4) << 1;
    for (int k0 = 0; k0 < kD; k0 += 4) {
      v2f a;
      a.x = S[m][k0 + koff];
      a.y = S[m][k0 + koff + 1];
      c = __builtin_amdgcn_wmma_f32_16x16x4_f32(false, a, false, a,
                                                (short)0, c, false, false);
    }
    // diagonal (m,m): m<8 -> lane m, vgpr m ; m>=8 -> lane m+16, vgpr m-8
    const float dv = c[lane & 7];
    if (lane < 8)        diag[lane]      = dv;
    else if (lane >= 24) diag[lane - 16] = dv;
#else
    if (lane < 16) {
      float s = 0.f;
      for (int d = 0; d < kD; ++d) s += S[lane][d] * S[lane][d];
      diag[lane] = s;
    }
#endif
  }
  __syncthreads();

  if (threadIdx.x == 0) {
    float r = 0.f;
    for (int b = 0; b < kB; ++b) r += diag[b];
    out[0] = r * (1.0f / ((float)kN * (float)kN * (float)kB));
  }
}

// =====================================================================
extern "C" void kernel_launch(void* const* d_in, const int* in_sizes, int n_in,
                              void* d_out, int out_size, void* d_ws, size_t ws_size,
                              hipStream_t stream)
{
  (void)in_sizes; (void)n_in; (void)out_size; (void)ws_size;
  const float* x       = (const float*)d_in[0];
  float*       partial = (float*)d_ws;   // 8 * 64 * 512 f32 = 1 MiB
  float*       out     = (float*)d_out;

  dim3 grid(kBlocksPerBatch, kB);
  assim_partials<<<grid, 256, 0, stream>>>(x, partial);
  assim_finalize<<<1, 256, 0, stream>>>(partial, out);
}